// PairInitOPM_3143916061385
// MI455X (gfx1250) — compile-verified
//
#include <hip/hip_runtime.h>
#include <hip/hip_bf16.h>

typedef float v2f __attribute__((ext_vector_type(2)));
typedef float v4f __attribute__((ext_vector_type(4)));
typedef float v8f __attribute__((ext_vector_type(8)));

#define L_SEQ 768
#define SD    384
#define PD    128

// ---------------------------------------------------------------------------
// Kernel 1: si = s @ wi^T + bi ; sj = s @ wj^T + bj
// One wave (32 lanes) per 16x16 output tile. 768 tiles total (48 M-tiles x
// 8 N-tiles x 2 matrices). f32 WMMA, K stepped by 4 (96 iterations).
//
// V_WMMA_F32_16X16X4_F32 operand layout (wave32):
//   A (16x4): lanes 0-15: M=lane, {v0,v1}={K,K+1}; lanes 16-31: {K+2,K+3}
//   B (4x16): lanes 0-15: N=lane, {v0,v1}={K,K+1}; lanes 16-31: {K+2,K+3}
//   C/D (16x16): VGPR r, lanes 0-15: M=r,N=lane; lanes 16-31: M=r+8
// ---------------------------------------------------------------------------
__global__ void __launch_bounds__(256)
proj_gemm_wmma(const float* __restrict__ s,
               const float* __restrict__ wi, const float* __restrict__ bi,
               const float* __restrict__ wj, const float* __restrict__ bj,
               float* __restrict__ si_out, float* __restrict__ sj_out) {
  const int wave = threadIdx.x >> 5;
  const int lane = threadIdx.x & 31;
  const int tile = blockIdx.x * 8 + wave;      // 0..767 (wave-uniform)
  const int mat  = (tile >= 384) ? 1 : 0;      // 0 = si, 1 = sj
  const int t    = mat ? tile - 384 : tile;
  const int mT   = t >> 3;                     // 0..47
  const int nT   = t & 7;                      // 0..7

  const float* w    = mat ? wj : wi;
  const float* bias = mat ? bj : bi;
  float*       out  = mat ? sj_out : si_out;

  const int lo = lane & 15;
  const int hi = lane >> 4;

  const float* ap = s + (mT * 16 + lo) * SD + 2 * hi;   // A: rows of s
  const float* bp = w + (nT * 16 + lo) * SD + 2 * hi;   // B: rows of w (= w^T cols)

  v8f c = {};
  #pragma unroll 4
  for (int k = 0; k < SD; k += 4) {
    v2f a = *(const v2f*)(ap + k);
    v2f b = *(const v2f*)(bp + k);
    c = __builtin_amdgcn_wmma_f32_16x16x4_f32(false, a, false, b,
                                              (short)0, c, false, false);
  }

  const int   n  = nT * 16 + lo;
  const float bb = bias[n];
  #pragma unroll
  for (int r = 0; r < 8; ++r) {
    const int m = mT * 16 + r + 8 * hi;
    out[m * PD + n] = c[r] + bb;
  }
}

// ---------------------------------------------------------------------------
// Kernel 2: zi = si @ wm[:, :PD]^T ; zj = sj @ wm[:, PD:]^T
// Same tiling; K = 128 (32 WMMA steps). wm row stride is 2*PD.
// ---------------------------------------------------------------------------
__global__ void __launch_bounds__(256)
mix_gemm_wmma(const float* __restrict__ si, const float* __restrict__ sj,
              const float* __restrict__ wm,
              float* __restrict__ zi, float* __restrict__ zj) {
  const int wave = threadIdx.x >> 5;
  const int lane = threadIdx.x & 31;
  const int tile = blockIdx.x * 8 + wave;      // 0..767
  const int mat  = (tile >= 384) ? 1 : 0;
  const int t    = mat ? tile - 384 : tile;
  const int mT   = t >> 3;
  const int nT   = t & 7;

  const float* A = mat ? sj : si;
  const float* W = wm + (mat ? PD : 0);        // column offset into wm
  float*     out = mat ? zj : zi;

  const int lo = lane & 15;
  const int hi = lane >> 4;

  const float* ap = A + (mT * 16 + lo) * PD + 2 * hi;
  const float* bp = W + (nT * 16 + lo) * (2 * PD) + 2 * hi;

  v8f c = {};
  #pragma unroll 4
  for (int k = 0; k < PD; k += 4) {
    v2f a = *(const v2f*)(ap + k);
    v2f b = *(const v2f*)(bp + k);
    c = __builtin_amdgcn_wmma_f32_16x16x4_f32(false, a, false, b,
                                              (short)0, c, false, false);
  }

  const int n = nT * 16 + lo;
  #pragma unroll
  for (int r = 0; r < 8; ++r) {
    const int m = mT * 16 + r + 8 * hi;
    out[m * PD + n] = c[r];
  }
}

// ---------------------------------------------------------------------------
// Kernel 3: out[i,j,e] = zi[i,e] + zj[j,e] + bm[e] + si[i,e]*sj[j,e]
// 302 MB streaming-store bound. Block = fixed i, 64-j chunk; 256 threads:
// tid&31 covers e in float4s (128 floats), tid>>5 picks j phase (8 rows in
// flight). Output written with nontemporal b128 stores (302 MB > 192 MB L2,
// keep sj/zj hot instead). sj/zj (768 KB) stay L2-resident across all i.
// ---------------------------------------------------------------------------
__global__ void __launch_bounds__(256)
pair_outer_bcast(const float* __restrict__ si, const float* __restrict__ sj,
                 const float* __restrict__ zi, const float* __restrict__ zj,
                 const float* __restrict__ bm, float* __restrict__ out) {
  const int i  = blockIdx.x;            // 0..767
  const int jc = blockIdx.y;            // 0..11 (chunks of 64 j's)
  const int e  = (threadIdx.x & 31) * 4;
  const int js = threadIdx.x >> 5;      // 0..7

  const v4f vsi  = *(const v4f*)(si + i * PD + e);
  const v4f vzi  = *(const v4f*)(zi + i * PD + e);
  const v4f vbm  = *(const v4f*)(bm + e);
  const v4f base = vzi + vbm;

  float* orow = out + (size_t)i * L_SEQ * PD;
  const int j0 = jc * 64;

  #pragma unroll 4
  for (int j = j0 + js; j < j0 + 64; j += 8) {
    const v4f vsj = *(const v4f*)(sj + j * PD + e);
    const v4f vzj = *(const v4f*)(zj + j * PD + e);
    const v4f r   = base + vzj + vsi * vsj;
    __builtin_nontemporal_store(r, (v4f*)(orow + (size_t)j * PD + e));
  }
}

// ---------------------------------------------------------------------------
extern "C" void kernel_launch(void* const* d_in, const int* in_sizes, int n_in,
                              void* d_out, int out_size, void* d_ws, size_t ws_size,
                              hipStream_t stream) {
  const float* s  = (const float*)d_in[0];  // [768, 384]
  const float* wi = (const float*)d_in[1];  // [128, 384]
  const float* bi = (const float*)d_in[2];  // [128]
  const float* wj = (const float*)d_in[3];  // [128, 384]
  const float* bj = (const float*)d_in[4];  // [128]
  const float* wm = (const float*)d_in[5];  // [128, 256]
  const float* bm = (const float*)d_in[6];  // [128]
  float* out = (float*)d_out;               // [768, 768, 128]

  const size_t MAT = (size_t)L_SEQ * PD;    // 98304 floats
  float* ws  = (float*)d_ws;
  float* si  = ws;
  float* sjv = ws + MAT;
  float* zi  = ws + 2 * MAT;
  float* zj  = ws + 3 * MAT;

  // 768 tiles / 8 waves per block = 96 blocks
  proj_gemm_wmma<<<96, 256, 0, stream>>>(s, wi, bi, wj, bj, si, sjv);
  mix_gemm_wmma<<<96, 256, 0, stream>>>(si, sjv, wm, zi, zj);

  dim3 grid(L_SEQ, L_SEQ / 64);             // (768, 12)
  pair_outer_bcast<<<grid, 256, 0, stream>>>(si, sjv, zi, zj, bm, out);
}